// MambaModel_12893491823417
// MI455X (gfx1250) — compile-verified
//
#include <hip/hip_runtime.h>
#include <math.h>

#define BB 32
#define TT 2048
#define FF 64
#define HH 512
#define SS 64
#define LL 4
#define BT (BB*TT)          // 65536 tokens
#define NTILES (BT/16)      // 4096 tiles of 16 tokens
#define LDA 516             // padded LDS stride (516 % 64banks = 4 -> conflict-free col reads)

typedef __attribute__((ext_vector_type(2))) float v2f;
typedef __attribute__((ext_vector_type(8))) float v8f;

// ---------------------------------------------------------------------------
// WMMA helpers: V_WMMA_F32_16X16X4_F32  (D = A(16x4) * B(4x16) + C(16x16))
// ---------------------------------------------------------------------------
__device__ __forceinline__ v8f wmma4(v8f acc, v2f a, v2f b) {
  // 8 args: (neg_a, A, neg_b, B, c_mod, C, reuse_a, reuse_b)
  return __builtin_amdgcn_wmma_f32_16x16x4_f32(false, a, false, b,
                                               (short)0, acc, false, false);
}

// A fragment: 16x4 tile from row-major src (leading dim lda), rows m=0..15,
// cols k0..k0+3. Lanes 0-15 hold K={k0,k0+1}, lanes 16-31 hold K={k0+2,k0+3}.
// Pair is contiguous -> single ds_load_b64 / global_load_b64.
__device__ __forceinline__ v2f frag_a(const float* src, int lda, int k0) {
  unsigned lane = threadIdx.x & 31u;
  int m = lane & 15;
  int k = k0 + ((lane >> 4) << 1);
  v2f r; r.x = src[m * lda + k]; r.y = src[m * lda + k + 1];
  return r;
}

// B fragment from PAIR-PACKED weights: element (k,n) of the logical KxN
// weight lives at Wp[(k>>1)*N + n] component (k&1). One b64 load per frag;
// K-loop advances the address by a constant 2*N*8 bytes (immediate offset).
__device__ __forceinline__ v2f frag_bp(const v2f* __restrict__ Wp, int N,
                                       int k0, int n0) {
  unsigned lane = threadIdx.x & 31u;
  int n = n0 + (lane & 15);
  int p = (k0 >> 1) + (lane >> 4);
  return Wp[(size_t)p * N + n];
}

// B fragment for W^T: logical element (k,n) stored at Wt[n*ldt + k]
// (contiguous in k -> merges to b64)
__device__ __forceinline__ v2f frag_bt(const float* Wt, int ldt, int k0, int n0) {
  unsigned lane = threadIdx.x & 31u;
  int n = n0 + (lane & 15);
  int k = k0 + ((lane >> 4) << 1);
  v2f r; r.x = Wt[n * ldt + k]; r.y = Wt[n * ldt + k + 1];
  return r;
}

__device__ __forceinline__ float sigmoidf_(float x) {
  return 1.f / (1.f + __expf(-x));
}

// ---------------------------------------------------------------------------
// Kernel 0: repack a KxN row-major weight into pair-packed fragment layout.
// Wp[((k>>1)*N + n)*2 + (k&1)] = W[k*N + n]
// ---------------------------------------------------------------------------
__global__ __launch_bounds__(256) void k_pack(
    const float* __restrict__ W, float* __restrict__ Wp, int K, int N) {
  int i = blockIdx.x * 256 + threadIdx.x;
  if (i >= K * N) return;
  int k = i / N, n = i - k * N;
  Wp[((size_t)(k >> 1) * N + n) * 2 + (k & 1)] = W[i];
}

// ---------------------------------------------------------------------------
// Kernel 1: h = x @ inW + inb      (BT x 64) @ (64 x 512)
// grid = NTILES, block = 128 (4 waves, 8 N-tiles each)
// ---------------------------------------------------------------------------
__global__ __launch_bounds__(128) void k_in_proj(
    const float* __restrict__ x, const float* __restrict__ inWp,
    const float* __restrict__ inb, float* __restrict__ h) {
  __shared__ float xs[16 * 68];
  long tok0 = (long)blockIdx.x * 16;
  for (int i = threadIdx.x; i < 16 * 64; i += 128)
    xs[(i >> 6) * 68 + (i & 63)] = x[tok0 * FF + i];
  __syncthreads();

  int wave = threadIdx.x >> 5;
  unsigned lane = threadIdx.x & 31u;
  const v2f* Wp = (const v2f*)inWp;
  for (int nt = 0; nt < 8; ++nt) {
    int n0 = (wave * 8 + nt) * 16;
    v8f acc = {0,0,0,0,0,0,0,0};
#pragma unroll
    for (int k0 = 0; k0 < FF; k0 += 4)
      acc = wmma4(acc, frag_a(xs, 68, k0), frag_bp(Wp, HH, k0, n0));
    int col = n0 + (lane & 15);
    int rbase = (lane >> 4) << 3;
    float bias = inb[col];
#pragma unroll
    for (int r = 0; r < 8; ++r)
      h[(tok0 + rbase + r) * HH + col] = acc[r] + bias;
  }
}

// ---------------------------------------------------------------------------
// Kernel 2: xn = LN(h);  xp = xn@xpW + xpb;  delta = sigmoid(xn@dpW + dpb)
// grid = NTILES, block = 256 (8 waves: LN 2 tokens/wave, then 1 tile/wave)
// ---------------------------------------------------------------------------
__global__ __launch_bounds__(256) void k_ln_proj(
    const float* __restrict__ h, const float* __restrict__ g,
    const float* __restrict__ bt, const float* __restrict__ xpWp,
    const float* __restrict__ xpb, const float* __restrict__ dpWp,
    const float* __restrict__ dpb, float* __restrict__ xp,
    float* __restrict__ dl) {
  __shared__ float xn[16 * LDA];
  long tok0 = (long)blockIdx.x * 16;
  int wave = threadIdx.x >> 5;
  unsigned lane = threadIdx.x & 31u;

  for (int i = 0; i < 2; ++i) {
    int tok = wave * 2 + i;
    const float* row = h + (tok0 + tok) * HH;
    float v[16];
    float s = 0.f, s2 = 0.f;
#pragma unroll
    for (int j = 0; j < 16; ++j) {
      float t = row[lane + 32 * j];
      v[j] = t; s += t; s2 += t * t;
    }
#pragma unroll
    for (int o = 16; o >= 1; o >>= 1) {
      s += __shfl_xor(s, o); s2 += __shfl_xor(s2, o);
    }
    float mean = s * (1.f / HH);
    float var = s2 * (1.f / HH) - mean * mean;
    float rstd = rsqrtf(var + 1e-5f);
#pragma unroll
    for (int j = 0; j < 16; ++j) {
      int c = lane + 32 * j;
      xn[tok * LDA + c] = (v[j] - mean) * rstd * g[c] + bt[c];
    }
  }
  __syncthreads();

  bool is_d = wave >= 4;
  const v2f* Wp = is_d ? (const v2f*)dpWp : (const v2f*)xpWp;
  const float* bb = is_d ? dpb : xpb;
  float* outp = is_d ? dl : xp;
  int n0 = (wave & 3) * 16;
  v8f acc = {0,0,0,0,0,0,0,0};
#pragma unroll 4
  for (int k0 = 0; k0 < HH; k0 += 4)
    acc = wmma4(acc, frag_a(xn, LDA, k0), frag_bp(Wp, SS, k0, n0));
  int col = n0 + (lane & 15);
  int rbase = (lane >> 4) << 3;
  float bias = bb[col];
#pragma unroll
  for (int r = 0; r < 8; ++r) {
    float val = acc[r] + bias;
    if (is_d) val = sigmoidf_(val);
    outp[(tok0 + rbase + r) * SS + col] = val;
  }
}

// ---------------------------------------------------------------------------
// Kernel 3: sequential scan.  state = (1-d)*state + d*(state@A^T + x_t)
// grid = B (32), block = S (64).  A-row register-resident, state via LDS.
// st_out may alias xp (read happens before write in the same thread).
// ---------------------------------------------------------------------------
__global__ __launch_bounds__(64) void k_scan(
    const float* __restrict__ xp, const float* __restrict__ dl,
    float* __restrict__ st_out, const float* __restrict__ A) {
  __shared__ float st[SS];
  int b = blockIdx.x;
  int s = threadIdx.x;
  float ar[SS];
#pragma unroll
  for (int j = 0; j < SS; ++j) ar[j] = A[s * SS + j];
  st[s] = 0.f;
  __syncthreads();
  long base = (long)b * TT * SS;
  for (int t = 0; t < TT; ++t) {
    long idx = base + (long)t * SS + s;
    float xv = xp[idx];
    float d = dl[idx];
    // pull next timestep's lines while we compute (global_prefetch_b8)
    __builtin_prefetch((const void*)(xp + idx + SS), 0, 0);
    __builtin_prefetch((const void*)(dl + idx + SS), 0, 0);
    float dot = 0.f;
#pragma unroll
    for (int j = 0; j < SS; ++j) dot += ar[j] * st[j];
    float ns = (1.f - d) * st[s] + d * (dot + xv);
    __syncthreads();
    st[s] = ns;
    st_out[idx] = ns;
    __syncthreads();
  }
}

// ---------------------------------------------------------------------------
// Kernel 4 (fused): xn = LN(h);  Y = states@C^T;  Y *= sigmoid(xn@gW+gb);
//                   h  = Y@oW + ob + h   (residual, in place)
// grid = NTILES, block = 512 (16 waves). Dynamic LDS: xn + yg (66 KB).
// ---------------------------------------------------------------------------
__global__ __launch_bounds__(512) void k_block_out(
    float* __restrict__ h, const float* __restrict__ g,
    const float* __restrict__ bt, const float* __restrict__ st,
    const float* __restrict__ Cw, const float* __restrict__ gWp,
    const float* __restrict__ gb, const float* __restrict__ oWp,
    const float* __restrict__ ob) {
  extern __shared__ float smem[];
  float* xn = smem;                 // 16 * LDA
  float* yg = smem + 16 * LDA;      // 16 * LDA
  long tok0 = (long)blockIdx.x * 16;
  int wave = threadIdx.x >> 5;
  unsigned lane = threadIdx.x & 31u;

  // Phase 1: LayerNorm, one wave per token
  {
    int tok = wave;
    const float* row = h + (tok0 + tok) * HH;
    float v[16];
    float s = 0.f, s2 = 0.f;
#pragma unroll
    for (int j = 0; j < 16; ++j) {
      float t = row[lane + 32 * j];
      v[j] = t; s += t; s2 += t * t;
    }
#pragma unroll
    for (int o = 16; o >= 1; o >>= 1) {
      s += __shfl_xor(s, o); s2 += __shfl_xor(s2, o);
    }
    float mean = s * (1.f / HH);
    float var = s2 * (1.f / HH) - mean * mean;
    float rstd = rsqrtf(var + 1e-5f);
#pragma unroll
    for (int j = 0; j < 16; ++j) {
      int c = lane + 32 * j;
      xn[tok * LDA + c] = (v[j] - mean) * rstd * g[c] + bt[c];
    }
  }
  __syncthreads();

  // Phase 2: Y = states(16x64) @ C^T  -> yg (each wave owns cols [32w,32w+32))
  const float* stt = st + tok0 * SS;
  for (int half = 0; half < 2; ++half) {
    int n0 = wave * 32 + half * 16;
    v8f acc = {0,0,0,0,0,0,0,0};
#pragma unroll
    for (int k0 = 0; k0 < SS; k0 += 4)
      acc = wmma4(acc, frag_a(stt, SS, k0), frag_bt(Cw, SS, k0, n0));
    int col = n0 + (lane & 15);
    int rbase = (lane >> 4) << 3;
#pragma unroll
    for (int r = 0; r < 8; ++r)
      yg[(rbase + r) * LDA + col] = acc[r];
  }

  // Phase 3: gate = sigmoid(xn@gW+gb); yg *= gate (same columns, same wave)
  {
    const v2f* Wp = (const v2f*)gWp;
    for (int half = 0; half < 2; ++half) {
      int n0 = wave * 32 + half * 16;
      v8f acc = {0,0,0,0,0,0,0,0};
#pragma unroll 4
      for (int k0 = 0; k0 < HH; k0 += 4) {
        if ((k0 & 63) == 0)  // stream packed weights 64 k-values ahead
          __builtin_prefetch(
              (const void*)(Wp + ((size_t)((k0 >> 1) + 32)) * HH + n0), 0, 1);
        acc = wmma4(acc, frag_a(xn, LDA, k0), frag_bp(Wp, HH, k0, n0));
      }
      int col = n0 + (lane & 15);
      int rbase = (lane >> 4) << 3;
      float bias = gb[col];
#pragma unroll
      for (int r = 0; r < 8; ++r)
        yg[(rbase + r) * LDA + col] *= sigmoidf_(acc[r] + bias);
    }
  }
  __syncthreads();

  // Phase 4: h = yg @ oW + ob + residual (in place)
  {
    const v2f* Wp = (const v2f*)oWp;
    for (int half = 0; half < 2; ++half) {
      int n0 = wave * 32 + half * 16;
      v8f acc = {0,0,0,0,0,0,0,0};
#pragma unroll 4
      for (int k0 = 0; k0 < HH; k0 += 4) {
        if ((k0 & 63) == 0)
          __builtin_prefetch(
              (const void*)(Wp + ((size_t)((k0 >> 1) + 32)) * HH + n0), 0, 1);
        acc = wmma4(acc, frag_a(yg, LDA, k0), frag_bp(Wp, HH, k0, n0));
      }
      int col = n0 + (lane & 15);
      int rbase = (lane >> 4) << 3;
      float bias = ob[col];
#pragma unroll
      for (int r = 0; r < 8; ++r) {
        long gi = (tok0 + rbase + r) * HH + col;
        h[gi] = acc[r] + bias + h[gi];
      }
    }
  }
}

// ---------------------------------------------------------------------------
// Kernel 5: head.  y = LN(h[:,T-1,:]); y = gelu(y@h1W+h1b); out = y@h2W+h2b
// Single block (32 rows). Uses global scratch for the 32x512 / 32x256 buffers.
// ---------------------------------------------------------------------------
__global__ __launch_bounds__(256) void k_head(
    const float* __restrict__ h, const float* __restrict__ hg,
    const float* __restrict__ hb, const float* __restrict__ h1W,
    const float* __restrict__ h1b, const float* __restrict__ h2W,
    const float* __restrict__ h2b, float* __restrict__ lnw,
    float* __restrict__ y1w, float* __restrict__ out) {
  __shared__ float red[32][16];
  int tid = threadIdx.x;
  int row = tid >> 3, part = tid & 7;
  const float* src = h + ((long)row * TT + (TT - 1)) * HH;

  float s = 0.f, s2 = 0.f;
  for (int j = 0; j < 64; ++j) {
    float t = src[part * 64 + j];
    s += t; s2 += t * t;
  }
  red[row][part] = s;
  red[row][part + 8] = s2;
  __syncthreads();
  if (part == 0) {
    float ss = 0.f, ss2 = 0.f;
    for (int j = 0; j < 8; ++j) { ss += red[row][j]; ss2 += red[row][j + 8]; }
    float mean = ss * (1.f / HH);
    float var = ss2 * (1.f / HH) - mean * mean;
    red[row][0] = mean;
    red[row][1] = rsqrtf(var + 1e-5f);
  }
  __syncthreads();
  float mean = red[row][0], rstd = red[row][1];
  for (int j = 0; j < 64; ++j) {
    int c = part * 64 + j;
    lnw[row * HH + c] = (src[c] - mean) * rstd * hg[c] + hb[c];
  }
  __syncthreads();

  // y1 (32 x 256): each thread computes 32 outputs
  for (int i = 0; i < 32; ++i) {
    int flat = tid * 32 + i;
    int b = flat >> 8, c = flat & 255;
    float a = h1b[c];
    for (int k = 0; k < HH; ++k) a += lnw[b * HH + k] * h1W[k * 256 + c];
    y1w[b * 256 + c] = 0.5f * a * (1.f + erff(a * 0.70710678118f));  // exact gelu
  }
  __syncthreads();

  if (tid < 32) {
    float a = h2b[0];
    for (int k = 0; k < 256; ++k) a += y1w[tid * 256 + k] * h2W[k];
    out[tid] = a;
  }
}

// ---------------------------------------------------------------------------
extern "C" void kernel_launch(void* const* d_in, const int* in_sizes, int n_in,
                              void* d_out, int out_size, void* d_ws,
                              size_t ws_size, hipStream_t stream) {
  const float* x    = (const float*)d_in[0];
  const float* inW  = (const float*)d_in[1];
  const float* inb  = (const float*)d_in[2];
  const float* ln_g = (const float*)d_in[3];
  const float* ln_b = (const float*)d_in[4];
  const float* xpW  = (const float*)d_in[5];
  const float* xpb  = (const float*)d_in[6];
  const float* dpW  = (const float*)d_in[7];
  const float* dpb  = (const float*)d_in[8];
  const float* Amat = (const float*)d_in[9];
  const float* Cmat = (const float*)d_in[10];
  const float* gW   = (const float*)d_in[11];
  const float* gb   = (const float*)d_in[12];
  const float* oW   = (const float*)d_in[13];
  const float* ob   = (const float*)d_in[14];
  const float* hg   = (const float*)d_in[15];
  const float* hb   = (const float*)d_in[16];
  const float* h1W  = (const float*)d_in[17];
  const float* h1b  = (const float*)d_in[18];
  const float* h2W  = (const float*)d_in[19];
  const float* h2b  = (const float*)d_in[20];

  const size_t SMALL = (size_t)HH * SS;   // 32768
  const size_t BIG   = (size_t)HH * HH;   // 262144

  float* h    = (float*)d_ws;                  // BT*HH   (128 MB)
  float* xp   = h + (size_t)BT * HH;           // BT*SS   (16 MB) — also states
  float* dl   = xp + (size_t)BT * SS;          // BT*SS   (16 MB)
  float* inWp = dl + (size_t)BT * SS;          // 64*512
  float* xpWp = inWp + (size_t)FF * HH;        // LL * 32768
  float* dpWp = xpWp + LL * SMALL;             // LL * 32768
  float* gWp  = dpWp + LL * SMALL;             // LL * 262144
  float* oWp  = gWp + LL * BIG;                // LL * 262144
  float* lnw  = oWp + LL * BIG;                // 32*512
  float* y1w  = lnw + (size_t)BB * HH;         // 32*256

  // --- pair-pack all weights into fragment-native layout (one b64/frag) ---
  k_pack<<<(FF * HH + 255) / 256, 256, 0, stream>>>(inW, inWp, FF, HH);
  for (int l = 0; l < LL; ++l) {
    k_pack<<<((int)SMALL + 255) / 256, 256, 0, stream>>>(
        xpW + l * SMALL, xpWp + l * SMALL, HH, SS);
    k_pack<<<((int)SMALL + 255) / 256, 256, 0, stream>>>(
        dpW + l * SMALL, dpWp + l * SMALL, HH, SS);
    k_pack<<<((int)BIG + 255) / 256, 256, 0, stream>>>(
        gW + l * BIG, gWp + l * BIG, HH, HH);
    k_pack<<<((int)BIG + 255) / 256, 256, 0, stream>>>(
        oW + l * BIG, oWp + l * BIG, HH, HH);
  }

  // --- model ---
  k_in_proj<<<NTILES, 128, 0, stream>>>(x, inWp, inb, h);
  for (int l = 0; l < LL; ++l) {
    k_ln_proj<<<NTILES, 256, 0, stream>>>(
        h, ln_g + l * HH, ln_b + l * HH, xpWp + l * SMALL, xpb + l * SS,
        dpWp + l * SMALL, dpb + l * SS, xp, dl);
    k_scan<<<BB, 64, 0, stream>>>(xp, dl, xp /*states in place*/,
                                  Amat + (size_t)l * SS * SS);
    k_block_out<<<NTILES, 512, 2 * 16 * LDA * sizeof(float), stream>>>(
        h, ln_g + l * HH, ln_b + l * HH, xp, Cmat + l * SMALL, gWp + l * BIG,
        gb + l * HH, oWp + l * BIG, ob + l * HH);
  }
  k_head<<<1, 256, 0, stream>>>(h, hg, hb, h1W, h1b, h2W, h2b, lnw, y1w,
                                (float*)d_out);
}